// GraphConvLayer_54657753809400
// MI455X (gfx1250) — compile-verified
//
#include <hip/hip_runtime.h>
#include <math.h>

// ---------------------------------------------------------------------------
// GraphConv layer:  out = gelu( BN(concat(x, x + ngh_sum)) @ W + b )
// N=8192 nodes, K=64 neighbors, W is [16384, 8192] fp32 (512 MB -> HBM bound).
// GEMV implemented with V_WMMA_F32_16X16X4_F32 (exact fp32), split-K = 16.
// ---------------------------------------------------------------------------

typedef __attribute__((ext_vector_type(2))) float v2f;
typedef __attribute__((ext_vector_type(8))) float v8f;

#define N_NODES 8192
#define K_NBRS  64
#define NROWS   16384          // 2*N
#define NCOLS   8192           // N
#define ROWS_PER_CHUNK 1024
#define NCHUNKS 16             // NROWS / ROWS_PER_CHUNK
#define BN_EPS  1e-3f

// --------------------------- kernel 1: build h -----------------------------
// one wave (32 lanes) per node: gather 64 neighbors, wave-reduce, apply BN.
__global__ __launch_bounds__(256) void build_h_kernel(
    const float* __restrict__ x, const int* __restrict__ adj,
    const float* __restrict__ gamma, const float* __restrict__ beta,
    const float* __restrict__ mean, const float* __restrict__ var,
    float* __restrict__ h) {
  const int node = blockIdx.x * 8 + (threadIdx.x >> 5);
  const int lane = threadIdx.x & 31;
  const int* arow = adj + node * K_NBRS;
  float s = x[arow[lane]] + x[arow[lane + 32]];
#pragma unroll
  for (int off = 16; off > 0; off >>= 1)
    s += __shfl_xor(s, off, 32);
  if (lane == 0) {
    const float xv  = x[node];
    const float agg = xv + s;                      // self + neighbor sum
    h[node] = (xv - mean[node]) * rsqrtf(var[node] + BN_EPS) * gamma[node]
              + beta[node];
    const int j2 = node + N_NODES;
    h[j2] = (agg - mean[j2]) * rsqrtf(var[j2] + BN_EPS) * gamma[j2]
            + beta[j2];
  }
}

// ------------------- kernel 2: split-K GEMV via WMMA -----------------------
// Per wave: 16 output columns, ROWS_PER_CHUNK rows of W.
// WMMA f32 16x16x4:
//   A[m,k] = h[r+k]  (broadcast over m)  -> one ds_load_b64 per step,
//            since A layout is K = vgpr + 2*(lane>=16), lane%16 = M.
//   B[k,n] = W[r+k, cbase+n]             -> two coalesced b32 loads
//            (vgpr v, lane L) = W[r + v + 2*(L>=16), cbase + L%16].
//   D[m,n] = partial dot for column cbase+n (identical across m).
__global__ __launch_bounds__(256) void gemv_wmma_kernel(
    const float* __restrict__ W, const float* __restrict__ h,
    float* __restrict__ partial) {
  __shared__ float hs[ROWS_PER_CHUNK];

  const int tid     = threadIdx.x;
  const int rowbase = blockIdx.y * ROWS_PER_CHUNK;

  // cooperative stage of this chunk's h into LDS (256 threads x float4)
  ((float4*)hs)[tid] = ((const float4*)(h + rowbase))[tid];
  __syncthreads();

  const int wave = tid >> 5;
  const int lane = tid & 31;
  const int hi   = lane >> 4;                       // half-wave select
  const int col  = blockIdx.x * 128 + wave * 16 + (lane & 15);

  // lane's W pointer: rows advance along K, column fixed by lane
  const float* p = W + (size_t)(rowbase + 2 * hi) * NCOLS + col;

  v8f acc = {0.f, 0.f, 0.f, 0.f, 0.f, 0.f, 0.f, 0.f};
#pragma unroll 4
  for (int r = 0; r < ROWS_PER_CHUNK; r += 4) {
    v2f a = *(const v2f*)(hs + r + 2 * hi);         // A: h[r+2hi], h[r+1+2hi]
    v2f bm;
    bm.x = p[0];                                    // W[r+0+2hi, col]
    bm.y = p[NCOLS];                                // W[r+1+2hi, col]
    p += 4 * NCOLS;
    acc = __builtin_amdgcn_wmma_f32_16x16x4_f32(
        /*neg_a=*/false, a, /*neg_b=*/false, bm,
        /*c_mod=*/(short)0, acc, /*reuse_a=*/false, /*reuse_b=*/false);
  }

  // every D row is identical; VGPR0 of lanes 0..15 holds (M=0, N=lane)
  if (lane < 16)
    partial[(size_t)blockIdx.y * NCOLS + col] = acc[0];
}

// ---------------- kernel 3: reduce split-K + bias + exact GELU -------------
__global__ __launch_bounds__(256) void finish_kernel(
    const float* __restrict__ partial, const float* __restrict__ bias,
    float* __restrict__ out) {
  const int j = blockIdx.x * 256 + threadIdx.x;
  float s = bias[j];
#pragma unroll
  for (int c = 0; c < NCHUNKS; ++c)                 // fixed order: deterministic
    s += partial[c * NCOLS + j];
  out[j] = 0.5f * s * (1.0f + erff(s * 0.70710678118654752f));
}

// ---------------------------------------------------------------------------
extern "C" void kernel_launch(void* const* d_in, const int* in_sizes, int n_in,
                              void* d_out, int out_size, void* d_ws, size_t ws_size,
                              hipStream_t stream) {
  const float* x     = (const float*)d_in[0];   // [1, N] fp32
  const int*   adj   = (const int*)d_in[1];     // [N, K] int
  // d_in[2] = edge_weights: unused by the reference
  const float* W     = (const float*)d_in[3];   // [2N, N] fp32
  const float* b     = (const float*)d_in[4];   // [N]
  const float* gamma = (const float*)d_in[5];   // [2N]
  const float* beta  = (const float*)d_in[6];   // [2N]
  const float* mean  = (const float*)d_in[7];   // [2N]
  const float* var   = (const float*)d_in[8];   // [2N]
  float* out = (float*)d_out;

  float* h       = (float*)d_ws;                // 16384 floats
  float* partial = h + NROWS;                   // NCHUNKS * 8192 floats

  build_h_kernel<<<N_NODES / 8, 256, 0, stream>>>(x, adj, gamma, beta, mean,
                                                  var, h);
  dim3 grid(NCOLS / 128, NCHUNKS);              // 64 col-tiles x 16 K-chunks
  gemv_wmma_kernel<<<grid, 256, 0, stream>>>(W, h, partial);
  finish_kernel<<<NCOLS / 256, 256, 0, stream>>>(partial, b, out);
}